// Simulation_22935125361184
// MI455X (gfx1250) — compile-verified
//
#include <hip/hip_runtime.h>
#include <stdint.h>

// ---------------------------------------------------------------------------
// Sequential 2-state double-integrator scan with teacher forcing + early stop.
// Nonlinear carried recurrence (division of state components) => no parallel
// scan possible. Optimal MI455X shape: one wave32; lanes cooperatively stage
// inputs into LDS via CDNA5 async global->LDS DMA (ASYNCcnt), lane 0 runs the
// latency-bound serial chain out of LDS with v_rcp_f32 on the critical path,
// then the wave drains outputs with async LDS->global stores.
// ---------------------------------------------------------------------------

#define N_STEPS_MAX 1001
#define DT_F        0.05f
#define HDT2_F      0.00125f   /* DT*DT/2 */
#define LO_BOUND    0.05f
#define HI_BOUND    1000.0f

__global__ __launch_bounds__(32)
void ctrl_scan_kernel(const float* __restrict__ inputs,
                      const float* __restrict__ fc1_w,
                      const int*   __restrict__ tf_mask,
                      float*       __restrict__ out,
                      int n)
{
    __shared__ float s_in[N_STEPS_MAX];
    __shared__ int   s_mask[N_STEPS_MAX];
    __shared__ float s_out[N_STEPS_MAX];

    const int t = threadIdx.x;
    if (n > N_STEPS_MAX) n = N_STEPS_MAX;

    // ---- Phase 1: async global -> LDS staging (CDNA5 ASYNCcnt path) ------
    // Low 32 bits of a flat shared-memory pointer are the LDS byte offset
    // (flat->LDS truncation rule, ISA 10.2).
    for (int i = t; i < n; i += 32) {
        unsigned           lds_in = (unsigned)(unsigned long long)&s_in[i];
        unsigned long long ga     = (unsigned long long)(inputs + i);
        asm volatile("global_load_async_to_lds_b32 %0, %1, off"
                     :: "v"(lds_in), "v"(ga) : "memory");
        unsigned           lds_mk = (unsigned)(unsigned long long)&s_mask[i];
        unsigned long long gm     = (unsigned long long)(tf_mask + i);
        asm volatile("global_load_async_to_lds_b32 %0, %1, off"
                     :: "v"(lds_mk), "v"(gm) : "memory");
    }
    asm volatile("s_wait_asynccnt 0" ::: "memory");
    __syncthreads();

    // ---- Phase 2: lane 0 runs the serial recurrence out of LDS -----------
    if (t == 0) {
        const float w = fc1_w[0];
        float s0 = 100.0f;   // X0
        float s1 = 0.0f;
        int   k_stop = n - 1;
        int   k = 0;
        for (; k < n; ++k) {
            const float u   = s_in[k];
            // teacher forcing: state_in = [s0, s0*u]; else [s0, s1]
            const float in1 = s_mask[k] ? (s0 * u) : s1;
            // net_in = u - in1/s0 ; net_out = w*net_in   (rcp: critical path)
            const float r   = in1 * __builtin_amdgcn_rcpf(s0);
            const float net = w * (u - r);
            // new_state = A@state_in + B*net
            const float ns0 = fmaf(HDT2_F, net, fmaf(DT_F, in1, s0));
            const float ns1 = fmaf(DT_F,  net, in1);
            // out = ns1/ns0 (off the carried chain; rcp is fine here)
            s_out[k] = ns1 * __builtin_amdgcn_rcpf(ns0);
            // stop check on the *new* state; hit step keeps its real output,
            // everything after is frozen at 0 (exact reference semantics)
            if (ns0 < LO_BOUND || ns0 > HI_BOUND) { k_stop = k; ++k; break; }
            s0 = ns0; s1 = ns1;
        }
        for (; k < n; ++k) s_out[k] = 0.0f;
        // k_stop is int32 in the reference tuple: store raw bits after outputs
        ((int*)out)[n] = k_stop;
    }
    __syncthreads();   // DS writes complete before async engine reads LDS

    // ---- Phase 3: async LDS -> global coalesced writeback ----------------
    for (int i = t; i < n; i += 32) {
        unsigned           lds_o = (unsigned)(unsigned long long)&s_out[i];
        unsigned long long ga    = (unsigned long long)(out + i);
        asm volatile("global_store_async_from_lds_b32 %0, %1, off"
                     :: "v"(ga), "v"(lds_o) : "memory");
    }
    asm volatile("s_wait_asynccnt 0" ::: "memory");
    // s_endpgm performs an implicit wait-idle as well.
}

extern "C" void kernel_launch(void* const* d_in, const int* in_sizes, int n_in,
                              void* d_out, int out_size, void* d_ws, size_t ws_size,
                              hipStream_t stream) {
    (void)n_in; (void)out_size; (void)d_ws; (void)ws_size;
    const float* inputs = (const float*)d_in[0];   // (1001,) f32
    const float* fc1_w  = (const float*)d_in[1];   // (1,1)   f32
    const int*   tf     = (const int*)d_in[2];     // (1001,) bool -> int
    float*       out    = (float*)d_out;           // 1001 f32 + 1 int32 (k_stop)
    const int n = in_sizes[0];
    // One wave32: the recurrence is inherently serial; more threads/blocks
    // cannot help and would only add launch/synchronization overhead.
    ctrl_scan_kernel<<<1, 32, 0, stream>>>(inputs, fc1_w, tf, out, n);
}